// CLDTransformer_45397804319399
// MI455X (gfx1250) — compile-verified
//
#include <hip/hip_runtime.h>
#include <math.h>

// ---------------------------------------------------------------------------
// CDNA5 / gfx1250 implementation of the CLD-Transformer reference.
// fp32 end-to-end using V_WMMA_F32_16X16X4_F32 (exact fp32 matrix cores).
// ---------------------------------------------------------------------------

typedef float v2f __attribute__((ext_vector_type(2)));
typedef float v8f __attribute__((ext_vector_type(8)));

#define WMMA_F32(a, b, c) \
  __builtin_amdgcn_wmma_f32_16x16x4_f32(false, (a), false, (b), (short)0, (c), false, false)

// ---- problem sizes --------------------------------------------------------
#define BATCH   256
#define CCH     4
#define TLEN    4096
#define PATCHW  16
#define PATCHN  256      // patches per sequence
#define HD      64
#define KCODE   128
#define NSEQ    1024     // B*C

// ---- workspace layout (float offsets) -------------------------------------
#define WS_TEMBM   0          // 64   mean_p temb
#define WS_TEMB0   64         // 64   temb at patch 0
#define WS_LAT     128        // 128*64  codebook @ W_lat + b_lat
#define WS_CBN     8320       // 128  |codebook_k|^2
#define WS_LOSSP   8448       // 1024 per-block VQ loss partials
#define WS_FEAT    9472       // 1024*64
#define WS_H0      75008      // 1024*64
#define WS_ADJ     140544     // 256*16
#define WS_POOLED  144640     // 256*64
// total 161024 floats = 644 KB

// ---- output layout ---------------------------------------------------------
#define OUT_LOGITS 0          // 256*2
#define OUT_FOCAL  512        // 256*4
#define OUT_TAU    1536       // 256*16
#define OUT_LOSS   5632       // commit, codebook

// ---------------------------------------------------------------------------
// Branch-free fast transcendentals: one v_exp_f32 + one v_rcp_f32, no exec
// divergence, saturates correctly at the tails (no NaN from inf/inf).
__device__ __forceinline__ float tanh_fast(float x) {
  float e = __expf(2.0f * x);            // v_exp_f32 (2^x after log2e scale)
  return 1.0f - 2.0f / (e + 1.0f);       // +inf -> 1, 0 -> -1
}
__device__ __forceinline__ float sigmoid_fast(float x) {
  return 1.0f / (1.0f + __expf(-x));
}
__device__ __forceinline__ float gelu_tanh(float x) {
  const float c0 = 0.7978845608028654f;  // sqrt(2/pi)
  const float c1 = 0.044715f;
  float x3 = x * x * x;
  return 0.5f * x * (1.0f + tanh_fast(c0 * (x + c1 * x3)));
}

__device__ __forceinline__ v8f zero8() {
  v8f z = {0.f, 0.f, 0.f, 0.f, 0.f, 0.f, 0.f, 0.f};
  return z;
}

// A fragment (16x4 fp32): lane l holds row (l&15), K = k0 + 2*(l>>4) .. +1
__device__ __forceinline__ v2f load_a_frag(const float* src, int ld, int k0, int lane) {
  int row = lane & 15;
  int kk  = k0 + ((lane >> 4) << 1);
  v2f a;
  a.x = src[row * ld + kk];
  a.y = src[row * ld + kk + 1];
  return a;
}
// B fragment (4x16 fp32) from row-major K x N storage
__device__ __forceinline__ v2f load_b_frag(const float* src, int ld, int k0, int n0, int lane) {
  int col = lane & 15;
  int kk  = k0 + ((lane >> 4) << 1);
  v2f b;
  b.x = src[kk * ld + n0 + col];
  b.y = src[(kk + 1) * ld + n0 + col];
  return b;
}
// B fragment where memory holds B^T (N x K row-major), i.e. B[k][n] = srcT[n][k]
__device__ __forceinline__ v2f load_bT_frag(const float* srcT, int ld, int k0, int n0, int lane) {
  int col = lane & 15;
  int kk  = k0 + ((lane >> 4) << 1);
  const float* p = srcT + (n0 + col) * ld + kk;
  v2f b;
  b.x = p[0];
  b.y = p[1];
  return b;
}

// ===========================================================================
// Kernel 0: temb_mean / temb0, lat_code = codebook@W_lat+b_lat, |cb|^2
// ===========================================================================
__global__ __launch_bounds__(256) void k_precompute(
    const float* __restrict__ codebook, const float* __restrict__ W_lat,
    const float* __restrict__ b_lat, const float* __restrict__ W_time,
    const float* __restrict__ b_time, float* __restrict__ ws) {
  __shared__ float meanbasis[64];
  int tid = threadIdx.x;

  if (tid < 64) {
    int j = tid & 31;
    bool iscos = tid >= 32;
    float freq = expf((float)j * (logf(1000.0f) / 31.0f));
    float acc = 0.f;
    for (int p = 0; p < PATCHN; ++p) {
      float t = (float)p * (16.0f / 255.0f);    // linspace(0, T/SR, 256)
      float a = t * freq;
      acc += iscos ? cosf(a) : sinf(a);
    }
    meanbasis[tid] = acc * (1.0f / 256.0f);
  }
  __syncthreads();

  if (tid < 64) {
    float am = 0.f, a0 = 0.f;
    for (int j = 0; j < 64; ++j) {
      float w = W_time[j * HD + tid];
      am += meanbasis[j] * w;
      if (j >= 32) a0 += w;                     // basis0: sin(0)=0, cos(0)=1
    }
    ws[WS_TEMBM + tid] = am + b_time[tid];
    ws[WS_TEMB0 + tid] = a0 + b_time[tid];
  }

  // lat_code
  for (int o = tid; o < KCODE * HD; o += 256) {
    int k = o >> 6, h = o & 63;
    float acc = b_lat[h];
    for (int d = 0; d < HD; ++d) acc += codebook[k * HD + d] * W_lat[d * HD + h];
    ws[WS_LAT + o] = acc;
  }
  // codebook norms
  for (int k = tid; k < KCODE; k += 256) {
    float s = 0.f;
    for (int d = 0; d < HD; ++d) { float v = codebook[k * HD + d]; s += v * v; }
    ws[WS_CBN + k] = s;
  }
}

// ===========================================================================
// Kernel 1: encoder + VQ.  One workgroup (4 waves) per (b,c) sequence.
//   Each wave processes 16-patch tiles with WMMA fp32 16x16x4 chains.
// ===========================================================================
__global__ __launch_bounds__(128) void k_encoder_vq(
    const float* __restrict__ x, const float* __restrict__ W_enc,
    const float* __restrict__ b_enc, const float* __restrict__ W_enc2,
    const float* __restrict__ b_enc2, const float* __restrict__ codebook,
    float* __restrict__ ws) {
  __shared__ float sWenc[PATCHW * HD];     //  4 KB
  __shared__ float sWenc2[HD * HD];        // 16 KB
  __shared__ float sCb[KCODE * HD];        // 32 KB (row-major codes x H = B^T)
  __shared__ float sLat[KCODE * HD];       // 32 KB
  __shared__ float sCbn[KCODE];
  __shared__ float sBenc[HD], sBenc2[HD];
  __shared__ float sZ1[4][PATCHW * HD];    // 16 KB (per wave)
  __shared__ float sZe[4][PATCHW * HD];    // 16 KB
  __shared__ float sNorm[4][16];
  __shared__ int   sIdx[4][16];
  __shared__ float sSumW[4][HD];
  __shared__ float sLossW[4];

  const int tid = threadIdx.x;
  const int lane = tid & 31;
  const int wv = tid >> 5;
  const int seq = blockIdx.x;

  for (int o = tid; o < PATCHW * HD; o += 128) sWenc[o] = W_enc[o];
  for (int o = tid; o < HD * HD; o += 128) sWenc2[o] = W_enc2[o];
  for (int o = tid; o < KCODE * HD; o += 128) {
    sCb[o] = codebook[o];
    sLat[o] = ws[WS_LAT + o];
  }
  for (int o = tid; o < KCODE; o += 128) sCbn[o] = ws[WS_CBN + o];
  if (tid < HD) { sBenc[tid] = b_enc[tid]; sBenc2[tid] = b_enc2[tid]; }
  __syncthreads();

  const float* xseq = x + (long)seq * TLEN;
  const int c0 = lane * 2, c1 = lane * 2 + 1;
  float latA0 = 0.f, latA1 = 0.f;   // per-lane feat accumulation (2 columns)
  float lossAcc = 0.f;              // full 16-patch min-distance sums

  for (int tile = wv; tile < 16; tile += 4) {
    const float* xt = xseq + tile * (PATCHW * PATCHW);   // 16 patches x 16

    // ---- z1 = gelu(x @ W_enc + b_enc) : 16x64, K=16 -----------------------
    #pragma unroll
    for (int nt = 0; nt < 4; ++nt) {
      v8f c = zero8();
      #pragma unroll
      for (int k0 = 0; k0 < PATCHW; k0 += 4) {
        v2f a = load_a_frag(xt, PATCHW, k0, lane);
        v2f b = load_b_frag(sWenc, HD, k0, nt * 16, lane);
        c = WMMA_F32(a, b, c);
      }
      int col = nt * 16 + (lane & 15);
      float bias = sBenc[col];
      #pragma unroll
      for (int i = 0; i < 8; ++i) {
        int m = (lane < 16) ? i : i + 8;
        sZ1[wv][m * HD + col] = gelu_tanh(c[i] + bias);
      }
    }
    __syncthreads();

    // ---- z_e = z1 @ W_enc2 + b_enc2 : 16x64, K=64 -------------------------
    #pragma unroll
    for (int nt = 0; nt < 4; ++nt) {
      v8f c = zero8();
      #pragma unroll
      for (int k0 = 0; k0 < HD; k0 += 4) {
        v2f a = load_a_frag(sZ1[wv], HD, k0, lane);
        v2f b = load_b_frag(sWenc2, HD, k0, nt * 16, lane);
        c = WMMA_F32(a, b, c);
      }
      int col = nt * 16 + (lane & 15);
      float bias = sBenc2[col];
      #pragma unroll
      for (int i = 0; i < 8; ++i) {
        int m = (lane < 16) ? i : i + 8;
        sZe[wv][m * HD + col] = c[i] + bias;
      }
    }
    __syncthreads();

    // ---- per-patch |z_e|^2 ------------------------------------------------
    if (lane < 16) {
      float s = 0.f;
      for (int h = 0; h < HD; ++h) { float v = sZe[wv][lane * HD + h]; s += v * v; }
      sNorm[wv][lane] = s;
    }
    __syncthreads();

    float rn[8];
    #pragma unroll
    for (int i = 0; i < 8; ++i) rn[i] = sNorm[wv][(lane < 16) ? i : i + 8];

    // ---- G = z_e @ codebook^T, argmin in C-fragment registers -------------
    float minv[8];
    int mini[8];
    #pragma unroll
    for (int i = 0; i < 8; ++i) { minv[i] = 3.4e38f; mini[i] = 0; }

    #pragma unroll
    for (int ntc = 0; ntc < 8; ++ntc) {
      v8f c = zero8();
      #pragma unroll
      for (int k0 = 0; k0 < HD; k0 += 4) {
        v2f a = load_a_frag(sZe[wv], HD, k0, lane);
        v2f b = load_bT_frag(sCb, HD, k0, ntc * 16, lane);
        c = WMMA_F32(a, b, c);
      }
      int n = ntc * 16 + (lane & 15);
      float cn = sCbn[n];
      #pragma unroll
      for (int i = 0; i < 8; ++i) {
        float d = rn[i] + cn - 2.0f * c[i];
        if (d < minv[i] || (d == minv[i] && n < mini[i])) { minv[i] = d; mini[i] = n; }
      }
    }
    // butterfly within each 16-lane half (rows differ between halves)
    #pragma unroll
    for (int off = 1; off < 16; off <<= 1) {
      #pragma unroll
      for (int i = 0; i < 8; ++i) {
        float ov = __shfl_xor(minv[i], off, 32);
        int oi = __shfl_xor(mini[i], off, 32);
        if (ov < minv[i] || (ov == minv[i] && oi < mini[i])) { minv[i] = ov; mini[i] = oi; }
      }
    }
    // all lanes of a half now agree; publish indices + loss
    if ((lane & 15) == 0) {
      #pragma unroll
      for (int i = 0; i < 8; ++i) sIdx[wv][((lane >> 4) << 3) + i] = mini[i];
    }
    float ls = 0.f;
    #pragma unroll
    for (int i = 0; i < 8; ++i) ls += minv[i];
    ls += __shfl_xor(ls, 16, 32);   // ls = full 16-patch sum, identical on all lanes
    lossAcc += ls;                  // lane 0 publishes once at the end
    __syncthreads();

    // ---- gather lat_code rows into feat accumulators ----------------------
    #pragma unroll 4
    for (int p = 0; p < 16; ++p) {
      int id = sIdx[wv][p];
      latA0 += sLat[id * HD + c0];
      latA1 += sLat[id * HD + c1];
    }
    if (tile == 0) {   // wave 0 only: h0 = lat_code[idx0] + temb0
      int id0 = sIdx[wv][0];
      ws[WS_H0 + seq * HD + c0] = sLat[id0 * HD + c0] + ws[WS_TEMB0 + c0];
      ws[WS_H0 + seq * HD + c1] = sLat[id0 * HD + c1] + ws[WS_TEMB0 + c1];
    }
    __syncthreads();
  }

  sSumW[wv][c0] = latA0;
  sSumW[wv][c1] = latA1;
  if (lane == 0) sLossW[wv] = lossAcc;
  __syncthreads();
  if (tid == 0) {
    ws[WS_LOSSP + blockIdx.x] = sLossW[0] + sLossW[1] + sLossW[2] + sLossW[3];
  }
  if (tid < HD) {
    float s = sSumW[0][tid] + sSumW[1][tid] + sSumW[2][tid] + sSumW[3][tid];
    ws[WS_FEAT + seq * HD + tid] = s * (1.0f / 256.0f) + ws[WS_TEMBM + tid];
  }
}

// ===========================================================================
// Kernel 2: edge MLP. One wave per batch b. Pair matrix is exactly 16x128.
// ===========================================================================
__global__ __launch_bounds__(32) void k_edge(
    const float* __restrict__ W_e1, const float* __restrict__ b_e1,
    const float* __restrict__ W_e2, const float* __restrict__ b_e2,
    float* __restrict__ ws, float* __restrict__ out) {
  __shared__ float sW[128 * HD];     // 32 KB
  __shared__ float sPair[16 * 128];  //  8 KB
  __shared__ float sHid[16 * HD];    //  4 KB
  __shared__ float sTau[16];

  const int b = blockIdx.x;
  const int lane = threadIdx.x;
  const float* feat = ws + WS_FEAT + (long)b * CCH * HD;

  for (int o = lane; o < 128 * HD; o += 32) sW[o] = W_e1[o];
  for (int o = lane; o < 16 * 128; o += 32) {
    int r = o >> 7, cn = o & 127;
    sPair[o] = (cn < HD) ? feat[(r >> 2) * HD + cn] : feat[(r & 3) * HD + (cn - HD)];
  }
  __syncthreads();

  #pragma unroll
  for (int nt = 0; nt < 4; ++nt) {
    v8f c = zero8();
    #pragma unroll
    for (int k0 = 0; k0 < 128; k0 += 4) {
      v2f a = load_a_frag(sPair, 128, k0, lane);
      v2f bb = load_b_frag(sW, HD, k0, nt * 16, lane);
      c = WMMA_F32(a, bb, c);
    }
    int col = nt * 16 + (lane & 15);
    float bias = b_e1[col];
    #pragma unroll
    for (int i = 0; i < 8; ++i) {
      int m = (lane < 16) ? i : i + 8;
      sHid[m * HD + col] = gelu_tanh(c[i] + bias);
    }
  }
  __syncthreads();

  const int r = lane & 15;   // pair index ci*4+cj
  const int j = lane >> 4;   // output column (0=edge, 1=tau)
  float e = b_e2[j];
  for (int h = 0; h < HD; ++h) e += sHid[r * HD + h] * W_e2[h * 2 + j];
  if (j == 1) {
    float tau = sigmoid_fast(e);
    out[OUT_TAU + b * 16 + r] = tau;
    sTau[r] = tau;
  } else {
    float p = sigmoid_fast(e);
    int ci = r >> 2, cj = r & 3;
    ws[WS_ADJ + b * 16 + r] = (ci != cj) ? p : 0.f;
  }
  __syncthreads();
  if (lane < 4) {
    float fs = sTau[lane * 4] + sTau[lane * 4 + 1] + sTau[lane * 4 + 2] + sTau[lane * 4 + 3];
    out[OUT_FOCAL + b * 4 + lane] = -fs;
  }
}

// ===========================================================================
// Kernel 3: RK4 neural ODE. One wave per block, 4 batches -> M=16 WMMA rows.
// ===========================================================================
template <int MODE>   // 0:k1  1:k2  2:k3  3:k4(final)
__device__ __forceinline__ void ode_stage(
    const float* sin_, float* snext, float* acc, float* s0, float* traj,
    const float* sWa, const float* sWs, const float* sBo,
    const float (&adjc)[8][4], int lane, float dt, float coef) {
  #pragma unroll
  for (int nt = 0; nt < 4; ++nt) {
    v8f cu = zero8();   // s @ W_adj
    v8f cv = zero8();   // s @ W_self
    #pragma unroll
    for (int k0 = 0; k0 < HD; k0 += 4) {
      v2f a = load_a_frag(sin_, HD, k0, lane);
      v2f bu = load_b_frag(sWa, HD, k0, nt * 16, lane);
      v2f bv = load_b_frag(sWs, HD, k0, nt * 16, lane);
      cu = WMMA_F32(a, bu, cu);
      cv = WMMA_F32(a, bv, cv);
    }
    int col = nt * 16 + (lane & 15);
    float bo = sBo[col];
    #pragma unroll
    for (int i = 0; i < 8; ++i) {
      const int q = i & 4;  // start of this VGPR's 4-row (channel) quad
      float mix = adjc[i][0] * cu[q + 0] + adjc[i][1] * cu[q + 1] +
                  adjc[i][2] * cu[q + 2] + adjc[i][3] * cu[q + 3];
      float kv = tanh_fast(mix + cv[i] + bo);
      int m = (lane < 16) ? i : i + 8;
      int off = m * HD + col;
      if (MODE == 0) acc[off] = kv;
      else if (MODE == 1 || MODE == 2) acc[off] += 2.0f * kv;
      if (MODE <= 2) {
        snext[off] = s0[off] + coef * dt * kv;
      } else {
        float at = acc[off] + kv;                 // k1+2k2+2k3+k4
        float sn = s0[off] + (dt / 6.0f) * at;
        s0[off] = sn;
        traj[off] += sn;
      }
    }
  }
}

__global__ __launch_bounds__(32) void k_ode(
    const float* __restrict__ W_adj, const float* __restrict__ W_self,
    const float* __restrict__ b_ode, float* __restrict__ ws) {
  __shared__ float sWa[HD * HD];      // 16 KB
  __shared__ float sWs[HD * HD];      // 16 KB
  __shared__ float sBo[HD];
  __shared__ float sS0[16 * HD];      //  4 KB each
  __shared__ float sStA[16 * HD];
  __shared__ float sStB[16 * HD];
  __shared__ float sAcc[16 * HD];
  __shared__ float sTraj[16 * HD];

  const int lane = threadIdx.x;
  const int bbase = blockIdx.x * 4;   // 4 batches per wave

  for (int o = lane; o < HD * HD; o += 32) { sWa[o] = W_adj[o]; sWs[o] = W_self[o]; }
  for (int o = lane; o < HD; o += 32) sBo[o] = b_ode[o];
  for (int o = lane; o < 16 * HD; o += 32) {
    float v = ws[WS_H0 + (bbase * 4 + (o >> 6)) * HD + (o & 63)];
    sS0[o] = v;
    sTraj[o] = v;                     // trajectory includes initial state
  }
  // per-lane adjacency coefficients for the two row-quads this lane half owns
  float adjc[8][4];
  #pragma unroll
  for (int i = 0; i < 8; ++i) {
    int lb = (i >> 2) + ((lane >> 4) << 1);   // local batch of row (i or i+8)
    #pragma unroll
    for (int j = 0; j < 4; ++j)
      adjc[i][j] = ws[WS_ADJ + (bbase + lb) * 16 + (i & 3) * 4 + j];
  }
  __syncthreads();

  const float dt = 16.0f / 255.0f;
  for (int step = 0; step < 255; ++step) {
    ode_stage<0>(sS0, sStA, sAcc, sS0, sTraj, sWa, sWs, sBo, adjc, lane, dt, 0.5f);
    __syncthreads();
    ode_stage<1>(sStA, sStB, sAcc, sS0, sTraj, sWa, sWs, sBo, adjc, lane, dt, 0.5f);
    __syncthreads();
    ode_stage<2>(sStB, sStA, sAcc, sS0, sTraj, sWa, sWs, sBo, adjc, lane, dt, 1.0f);
    __syncthreads();
    ode_stage<3>(sStA, sStB, sAcc, sS0, sTraj, sWa, sWs, sBo, adjc, lane, dt, 0.f);
    __syncthreads();
  }

  // pooled_raw[b][h] = sum over channels & time (divide later)
  for (int o = lane; o < 4 * HD; o += 32) {
    int lb = o >> 6, h = o & 63;
    float s = sTraj[(lb * 4 + 0) * HD + h] + sTraj[(lb * 4 + 1) * HD + h] +
              sTraj[(lb * 4 + 2) * HD + h] + sTraj[(lb * 4 + 3) * HD + h];
    ws[WS_POOLED + (bbase + lb) * HD + h] = s;
  }
}

// ===========================================================================
// Kernel 4: LayerNorm + MLP head + loss finalize
// ===========================================================================
__global__ __launch_bounds__(64) void k_head(
    const float* __restrict__ ln_g, const float* __restrict__ ln_b,
    const float* __restrict__ W_h1, const float* __restrict__ b_h1,
    const float* __restrict__ W_h2, const float* __restrict__ b_h2,
    const float* __restrict__ ws, float* __restrict__ out) {
  __shared__ float red[64];
  __shared__ float sN[64];
  __shared__ float sHid[64];
  const int b = blockIdx.x;
  const int h = threadIdx.x;

  float xv = ws[WS_POOLED + b * HD + h] * (1.0f / 1024.0f);   // /(256 steps * 4 ch)
  red[h] = xv;
  __syncthreads();
  for (int s = 32; s > 0; s >>= 1) { if (h < s) red[h] += red[h + s]; __syncthreads(); }
  float mu = red[0] * (1.0f / 64.0f);
  __syncthreads();
  float d = xv - mu;
  red[h] = d * d;
  __syncthreads();
  for (int s = 32; s > 0; s >>= 1) { if (h < s) red[h] += red[h + s]; __syncthreads(); }
  float var = red[0] * (1.0f / 64.0f);
  __syncthreads();

  sN[h] = d * rsqrtf(var + 1e-5f) * ln_g[h] + ln_b[h];
  __syncthreads();
  float acc = b_h1[h];
  for (int k = 0; k < HD; ++k) acc += sN[k] * W_h1[k * HD + h];
  sHid[h] = gelu_tanh(acc);
  __syncthreads();
  if (h < 2) {
    float a = b_h2[h];
    for (int k = 0; k < HD; ++k) a += sHid[k] * W_h2[k * 2 + h];
    out[OUT_LOGITS + b * 2 + h] = a;
  }

  if (b == 0) {   // block-uniform: reduce the 1024 VQ-loss partials
    __syncthreads();
    float lp = 0.f;
    for (int k = 0; k < 16; ++k) lp += ws[WS_LOSSP + h * 16 + k];
    red[h] = lp;
    __syncthreads();
    for (int s = 32; s > 0; s >>= 1) { if (h < s) red[h] += red[h + s]; __syncthreads(); }
    if (h == 0) {
      float L = red[0] * (1.0f / 16777216.0f);  // mean over 1024*256*64 elements
      out[OUT_LOSS + 0] = L;                    // commit_loss
      out[OUT_LOSS + 1] = L;                    // codebook_loss (same fwd value)
    }
  }
}

// ===========================================================================
extern "C" void kernel_launch(void* const* d_in, const int* in_sizes, int n_in,
                              void* d_out, int out_size, void* d_ws, size_t ws_size,
                              hipStream_t stream) {
  const float* x        = (const float*)d_in[0];
  const float* codebook = (const float*)d_in[1];
  const float* W_enc    = (const float*)d_in[2];
  const float* b_enc    = (const float*)d_in[3];
  const float* W_enc2   = (const float*)d_in[4];
  const float* b_enc2   = (const float*)d_in[5];
  const float* W_lat    = (const float*)d_in[6];
  const float* b_lat    = (const float*)d_in[7];
  const float* W_time   = (const float*)d_in[8];
  const float* b_time   = (const float*)d_in[9];
  const float* W_e1     = (const float*)d_in[10];
  const float* b_e1     = (const float*)d_in[11];
  const float* W_e2     = (const float*)d_in[12];
  const float* b_e2     = (const float*)d_in[13];
  const float* W_adj    = (const float*)d_in[14];
  const float* W_self   = (const float*)d_in[15];
  const float* b_ode    = (const float*)d_in[16];
  const float* ln_g     = (const float*)d_in[17];
  const float* ln_b     = (const float*)d_in[18];
  const float* W_h1     = (const float*)d_in[19];
  const float* b_h1     = (const float*)d_in[20];
  const float* W_h2     = (const float*)d_in[21];
  const float* b_h2     = (const float*)d_in[22];
  float* out = (float*)d_out;
  float* ws  = (float*)d_ws;
  (void)in_sizes; (void)n_in; (void)out_size; (void)ws_size;

  k_precompute<<<1, 256, 0, stream>>>(codebook, W_lat, b_lat, W_time, b_time, ws);
  k_encoder_vq<<<NSEQ, 128, 0, stream>>>(x, W_enc, b_enc, W_enc2, b_enc2, codebook, ws);
  k_edge<<<BATCH, 32, 0, stream>>>(W_e1, b_e1, W_e2, b_e2, ws, out);
  k_ode<<<BATCH / 4, 32, 0, stream>>>(W_adj, W_self, b_ode, ws);
  k_head<<<BATCH, 64, 0, stream>>>(ln_g, ln_b, W_h1, b_h1, W_h2, b_h2, ws, out);
}